// mIoU_37855841747376
// MI455X (gfx1250) — compile-verified
//
#include <hip/hip_runtime.h>
#include <hip/hip_bf16.h>
#include <cfloat>
#include <cstdint>

// ShapeNet class -> contiguous part-id range starts
__constant__ int d_starts[17] = {0, 4, 6, 8, 12, 16, 19, 22, 24, 28, 30, 36, 38, 41, 44, 47, 50};

#define NCHUNK 1024   // points per block
#define MAXP   6      // max parts per class (range 30..36)

// ---------------------------------------------------------------------------
// Kernel 1: masked argmax + per-(b, part) histograms.
// grid = (N/NCHUNK, B): block (x,y) handles points [x*NCHUNK, ...) of sample y.
// The <=6 masked logit rows are staged into LDS with CDNA5 async global->LDS
// DMA (ASYNCcnt-tracked), then each thread argmaxes 4 points out of LDS.
// pred_cnt / inter histograms use wave32 ballots (1 scalar LDS atomic per
// wave per bin); gt_cnt (50 spread-out bins) uses per-lane LDS atomics.
// ---------------------------------------------------------------------------
__global__ __launch_bounds__(256)
void argmax_hist_kernel(const float* __restrict__ logits,
                        const long long* __restrict__ targets,
                        const long long* __restrict__ labels,
                        void* __restrict__ pred_out, int pred_is_64,
                        unsigned* __restrict__ ws,
                        int B, int C, int N)
{
    __shared__ __attribute__((aligned(16))) float tile[MAXP * NCHUNK];
    __shared__ unsigned s_inter[64];
    __shared__ unsigned s_pcnt[64];
    __shared__ unsigned s_gcnt[64];

    const int t  = threadIdx.x;
    const int b  = blockIdx.y;
    const int n0 = blockIdx.x * NCHUNK;

    const int cls   = (int)labels[b];
    const int c0    = d_starts[cls];
    const int npart = d_starts[cls + 1] - c0;

    if (t < 64) { s_inter[t] = 0u; s_pcnt[t] = 0u; s_gcnt[t] = 0u; }

    const int nrem = min(NCHUNK, N - n0);
    if (nrem == NCHUNK) {
        // Async DMA: each thread moves one float4 per masked row.
        // 256 threads x 16B = one full 4KB row per iteration.
        for (int r = 0; r < npart; ++r) {
            const float* gp = logits + ((size_t)b * C + (size_t)(c0 + r)) * (size_t)N
                                     + (size_t)n0 + (size_t)(4 * t);
            unsigned           lds_addr = (unsigned)(uintptr_t)(&tile[r * NCHUNK + 4 * t]);
            unsigned long long gaddr    = (unsigned long long)(uintptr_t)gp;
            asm volatile("global_load_async_to_lds_b128 %0, %1, off"
                         :: "v"(lds_addr), "v"(gaddr) : "memory");
        }
        asm volatile("s_wait_asynccnt 0x0" ::: "memory");
    } else {
        // Tail chunk (not hit for N=65536): guarded direct loads.
        for (int r = 0; r < npart; ++r)
            for (int i = t; i < NCHUNK; i += 256)
                tile[r * NCHUNK + i] =
                    (i < nrem) ? logits[((size_t)b * C + (size_t)(c0 + r)) * (size_t)N
                                        + (size_t)n0 + (size_t)i]
                               : -FLT_MAX;
    }
    __syncthreads();

    const int BC   = B * C;
    const int lane = t & 31;

    #pragma unroll
    for (int i = 0; i < NCHUNK / 256; ++i) {
        const int  nl     = i * 256 + t;
        const bool active = (nl < nrem);

        // Argmax over the masked, contiguous part range (first-max wins, as in
        // jnp.argmax: strict > keeps the lowest index on ties).
        float best = active ? tile[nl] : -FLT_MAX;
        int   bc   = 0;
        for (int r = 1; r < npart; ++r) {
            const float v = active ? tile[r * NCHUNK + nl] : -FLT_MAX;
            if (v > best) { best = v; bc = r; }
        }

        int tg = -1;
        if (active) {
            const int       predc  = c0 + bc;
            const long long n_glob = (long long)b * N + n0 + nl;
            if (pred_is_64) ((long long*)pred_out)[n_glob] = (long long)predc;
            else            ((int*)pred_out)[n_glob]       = predc;
            tg = (int)targets[n_glob];
        }
        const bool match = active && ((c0 + bc) == tg);

        // Wave-aggregated pred/inter histogram: <=6 bins per class.
        for (int r = 0; r < npart; ++r) {
            const unsigned long long mp = __ballot(active && (bc == r));
            const unsigned long long mi = __ballot(match && (bc == r));
            if (lane == 0) {
                if (mp) atomicAdd(&s_pcnt[c0 + r],  (unsigned)__popcll(mp));
                if (mi) atomicAdd(&s_inter[c0 + r], (unsigned)__popcll(mi));
            }
        }
        // gt histogram: 50 bins spread across LDS banks -> per-lane atomics.
        if (active) atomicAdd(&s_gcnt[tg], 1u);
    }
    __syncthreads();

    if (t < C) {
        if (s_inter[t]) atomicAdd(&ws[(size_t)b * C + t],                  s_inter[t]);
        if (s_pcnt[t])  atomicAdd(&ws[(size_t)BC + (size_t)b * C + t],     s_pcnt[t]);
        if (s_gcnt[t])  atomicAdd(&ws[2 * (size_t)BC + (size_t)b * C + t], s_gcnt[t]);
    }
}

// ---------------------------------------------------------------------------
// Kernel 2: IoU per (b, masked part) -> masked mean per b -> mean over b.
// Tiny reduction: one block, thread b handles sample b.
// ---------------------------------------------------------------------------
__global__ __launch_bounds__(256)
void iou_finalize_kernel(const unsigned* __restrict__ ws,
                         const long long* __restrict__ labels,
                         float* __restrict__ out, int B, int C)
{
    __shared__ float s[256];
    const int b = threadIdx.x;
    float per = 0.f;
    if (b < B) {
        const int cls = (int)labels[b];
        const int c0  = d_starts[cls];
        const int c1  = d_starts[cls + 1];
        const int BC  = B * C;
        float sum = 0.f;
        for (int c = c0; c < c1; ++c) {
            const unsigned inter = ws[(size_t)b * C + c];
            const unsigned pc    = ws[(size_t)BC + (size_t)b * C + c];
            const unsigned gc    = ws[2 * (size_t)BC + (size_t)b * C + c];
            const unsigned un    = pc + gc - inter;
            sum += (un == 0u) ? 1.0f : (float)inter / (float)un;
        }
        per = sum / (float)(c1 - c0);
    }
    s[threadIdx.x] = per;
    __syncthreads();
    if (threadIdx.x == 0) {
        float tot = 0.f;
        for (int i = 0; i < B; ++i) tot += s[i];
        out[0] = tot / (float)B;
    }
}

extern "C" void kernel_launch(void* const* d_in, const int* in_sizes, int n_in,
                              void* d_out, int out_size, void* d_ws, size_t ws_size,
                              hipStream_t stream)
{
    const float*     logits  = (const float*)d_in[0];
    const long long* targets = (const long long*)d_in[1];
    const long long* labels  = (const long long*)d_in[2];

    const int B  = in_sizes[2];
    const int N  = in_sizes[1] / B;
    const int C  = (int)((long long)in_sizes[0] / ((long long)B * (long long)N));
    const int BC = B * C;

    unsigned* ws = (unsigned*)d_ws;
    hipMemsetAsync(ws, 0, (size_t)3 * (size_t)BC * sizeof(unsigned), stream);

    float* out      = (float*)d_out;
    void*  pred_out = (void*)(out + 1);
    // pred dtype ambiguity (int32 vs int64 under jax x64): decide from out_size.
    const long long need64 = 1LL + 2LL * (long long)B * (long long)N;
    const int pred_is_64   = ((long long)out_size >= need64) ? 1 : 0;

    const int chunksPerB = (N + NCHUNK - 1) / NCHUNK;
    dim3 grid((unsigned)chunksPerB, (unsigned)B);
    argmax_hist_kernel<<<grid, 256, 0, stream>>>(logits, targets, labels,
                                                 pred_out, pred_is_64, ws,
                                                 B, C, N);
    iou_finalize_kernel<<<1, 256, 0, stream>>>(ws, labels, out, B, C);
}